// Watson_16707422782107
// MI455X (gfx1250) — compile-verified
//
#include <hip/hip_runtime.h>
#include <math.h>

typedef float v2f __attribute__((ext_vector_type(2)));
typedef float v8f __attribute__((ext_vector_type(8)));

#define NP 256            // P (feature dim)
#define NK 64             // K (mixture components)
#define ROWS_PER_BLOCK 256
#define THREADS 256
#define A_PARAM 0.5f
#define C_PARAM 128.0f    // P/2

// ---------------------------------------------------------------------------
// Kernel 1: tiny prep (1 block).  mu -> unit columns, transposed to mu_t[k][p];
// per-component fused coefficients c0[k], c1[k].
// ---------------------------------------------------------------------------
__global__ __launch_bounds__(THREADS) void watson_prep(
    const float* __restrict__ mu, const float* __restrict__ kappa,
    const float* __restrict__ pi, float* __restrict__ mu_t,
    float* __restrict__ c0, float* __restrict__ c1)
{
  __shared__ float rnorm[NK];
  const int tid = threadIdx.x;

  if (tid < NK) {
    float s = 0.f;
    for (int p = 0; p < NP; ++p) { float v = mu[p * NK + tid]; s += v * v; }
    rnorm[tid] = 1.0f / sqrtf(s);
  }
  __syncthreads();

  // mu_t[k*NP + p] = mu[p*NK + k] / ||mu_k||   (contiguous along p)
  for (int i = tid; i < NP * NK; i += THREADS) {
    int p = i >> 6, k = i & (NK - 1);
    mu_t[k * NP + p] = mu[i] * rnorm[k];
  }

  if (tid < NK) {
    // log_softmax(pi)[tid]  (serial over 64 — trivial)
    float m = -3.4e38f;
    for (int j = 0; j < NK; ++j) m = fmaxf(m, pi[j]);
    float se = 0.f;
    for (int j = 0; j < NK; ++j) se += expf(pi[j] - m);
    float lpi = pi[tid] - (m + logf(se));

    // kappa_pos = softplus(kappa)
    float x  = kappa[tid];
    float kp = (x > 20.f) ? x : log1pf(expf(x));

    // log M(a,c,kp) series via logaddexp, mirrors reference while-loop
    float logkum = 0.f, old = 1.f, foo = 0.f;
    for (int j = 1; j < 10000; ++j) {
      if (fabsf(logkum - old) <= 1e-10f) break;
      float jf = (float)j;
      foo += logf((A_PARAM + jf - 1.f) / (jf * (C_PARAM + jf - 1.f)) * kp);
      float mx = fmaxf(logkum, foo);
      float nv = mx + log1pf(expf(fminf(logkum, foo) - mx));
      old = logkum; logkum = nv;
    }

    float logSA = lgammaf(C_PARAM) - logf(2.0f)
                - C_PARAM * logf(3.14159265358979323846f);
    c0[tid] = logSA - logkum + lpi;   // density = c0 + c1 * proj^2
    c1[tid] = kp;
  }
}

// ---------------------------------------------------------------------------
// Kernel 2: main GEMM (fp32 WMMA 16x16x4) + fused logsumexp epilogue.
// 8 waves/block; each wave owns TWO 16-row M-tiles (32 rows) x all 64
// components (4 N-tiles) => 8 accumulator tiles, 8 WMMAs per k-step while
// the 4 B fragments are shared between both M-tiles.
// A-fragment loads are unconditional (row clamped, masked by 0/1) and
// software-pipelined one k-step ahead.
// ---------------------------------------------------------------------------
__global__ __launch_bounds__(THREADS) void watson_main(
    const float* __restrict__ X, const float* __restrict__ mu_t,
    const float* __restrict__ c0, const float* __restrict__ c1,
    float* __restrict__ partials, int n_rows)
{
  __shared__ float smem[NP * NK + 2 * NK + THREADS];
  float* s_mu  = smem;                 // 16384 floats (64 KB): mu_t staged
  float* s_c0  = smem + NP * NK;       // 64
  float* s_c1  = s_c0 + NK;            // 64
  float* s_red = s_c1 + NK;            // 256

  const int tid = threadIdx.x;

  // Stage mu_t into LDS (coalesced float4 copies), plus coefficients.
  {
    const float4* src4 = (const float4*)mu_t;
    float4*       dst4 = (float4*)s_mu;
    for (int i = tid; i < (NP * NK) / 4; i += THREADS) dst4[i] = src4[i];
    if (tid < NK) { s_c0[tid] = c0[tid]; s_c1[tid] = c1[tid]; }
  }
  __syncthreads();

  const int lane = tid & 31;
  const int wid  = tid >> 5;
  const int half = lane >> 4;   // 0: lanes 0-15, 1: lanes 16-31
  const int ln   = lane & 15;

  const int rowbase = blockIdx.x * ROWS_PER_BLOCK + wid * 32;
  const int r0 = rowbase + ln;          // M-tile 0 row (M = ln)
  const int r1 = rowbase + 16 + ln;     // M-tile 1 row
  const float msk0 = (r0 < n_rows) ? 1.f : 0.f;
  const float msk1 = (r1 < n_rows) ? 1.f : 0.f;
  const float* x0 = X + (size_t)((r0 < n_rows) ? r0 : (n_rows - 1)) * NP;
  const float* x1 = X + (size_t)((r1 < n_rows) ? r1 : (n_rows - 1)) * NP;

  const v8f vzero = {0.f, 0.f, 0.f, 0.f, 0.f, 0.f, 0.f, 0.f};
  v8f acc0[4] = {vzero, vzero, vzero, vzero};   // M-tile 0, N-tiles 0..3
  v8f acc1[4] = {vzero, vzero, vzero, vzero};   // M-tile 1, N-tiles 0..3

  // P=256 contraction in K=4 steps, A fragments pipelined one step ahead.
  // A (16x4 f32): lane = row M=ln; VGPR pair = cols {4s+2h, 4s+2h+1}.
  // B (4x16 f32): lane = col N=ln; VGPR pair = rows {4s+2h, 4s+2h+1}.
  v2f a0, a1;
  {
    const int p0 = 2 * half;
    float2 t0 = *(const float2*)(x0 + p0);
    float2 t1 = *(const float2*)(x1 + p0);
    a0.x = t0.x * msk0; a0.y = t0.y * msk0;
    a1.x = t1.x * msk1; a1.y = t1.y * msk1;
  }
  for (int s = 0; s < NP / 4; ++s) {
    // prefetch next step's A fragments (wraps to step 0 on last iter; benign)
    const int pn = 4 * ((s + 1) & (NP / 4 - 1)) + 2 * half;
    const float2 t0 = *(const float2*)(x0 + pn);
    const float2 t1 = *(const float2*)(x1 + pn);

    const int p0 = 4 * s + 2 * half;
#pragma unroll
    for (int kt = 0; kt < 4; ++kt) {
      float2 bv = *(const float2*)(s_mu + (kt * 16 + ln) * NP + p0);
      v2f b; b.x = bv.x; b.y = bv.y;
      acc0[kt] = __builtin_amdgcn_wmma_f32_16x16x4_f32(
          false, a0, false, b, (short)0, acc0[kt], false, false);
      acc1[kt] = __builtin_amdgcn_wmma_f32_16x16x4_f32(
          false, a1, false, b, (short)0, acc1[kt], false, false);
    }

    a0.x = t0.x * msk0; a0.y = t0.y * msk0;
    a1.x = t1.x * msk1; a1.y = t1.y * msk1;
  }

  // Epilogue: density = c0[k] + c1[k]*proj^2; logsumexp over 64 comps/row.
  // C-tile layout: acc{t}[kt][j] @ lane = proj[rowbase + t*16 + j + 8*half]
  //                                           [comp = kt*16 + ln]
  float c0l[4], c1l[4];
#pragma unroll
  for (int kt = 0; kt < 4; ++kt) {
    c0l[kt] = s_c0[kt * 16 + ln];
    c1l[kt] = s_c1[kt * 16 + ln];
  }

  float local = 0.f;
#pragma unroll
  for (int t = 0; t < 2; ++t) {
    const v8f* acc = t ? acc1 : acc0;
#pragma unroll
    for (int j = 0; j < 8; ++j) {
      float d[4];
#pragma unroll
      for (int kt = 0; kt < 4; ++kt) {
        float pv = acc[kt][j];
        d[kt] = fmaf(c1l[kt], pv * pv, c0l[kt]);
      }
      // max over the row's 64 components: 4 local + 16-lane xor reduction
      float m = fmaxf(fmaxf(d[0], d[1]), fmaxf(d[2], d[3]));
#pragma unroll
      for (int off = 1; off < 16; off <<= 1)
        m = fmaxf(m, __shfl_xor(m, off, 32));
      float ss = expf(d[0] - m) + expf(d[1] - m)
               + expf(d[2] - m) + expf(d[3] - m);
#pragma unroll
      for (int off = 1; off < 16; off <<= 1)
        ss += __shfl_xor(ss, off, 32);
      if (ln == 0) {
        int r = rowbase + t * 16 + j + 8 * half;
        if (r < n_rows) local += m + logf(ss);
      }
    }
  }

  // Deterministic block tree-reduction -> one partial per block.
  s_red[tid] = local;
  __syncthreads();
  for (int st = THREADS / 2; st > 0; st >>= 1) {
    if (tid < st) s_red[tid] += s_red[tid + st];
    __syncthreads();
  }
  if (tid == 0) partials[blockIdx.x] = s_red[0];
}

// ---------------------------------------------------------------------------
// Kernel 3: deterministic final reduction of per-block partials.
// ---------------------------------------------------------------------------
__global__ __launch_bounds__(THREADS) void watson_reduce(
    const float* __restrict__ partials, int nblocks, float* __restrict__ out)
{
  __shared__ float s[THREADS];
  float a = 0.f;
  for (int i = threadIdx.x; i < nblocks; i += THREADS) a += partials[i];
  s[threadIdx.x] = a;
  __syncthreads();
  for (int st = THREADS / 2; st > 0; st >>= 1) {
    if (threadIdx.x < st) s[threadIdx.x] += s[threadIdx.x + st];
    __syncthreads();
  }
  if (threadIdx.x == 0) out[0] = s[0];
}

// ---------------------------------------------------------------------------
extern "C" void kernel_launch(void* const* d_in, const int* in_sizes, int n_in,
                              void* d_out, int out_size, void* d_ws, size_t ws_size,
                              hipStream_t stream) {
  const float* X     = (const float*)d_in[0];   // (N, 256)
  const float* mu    = (const float*)d_in[1];   // (256, 64)
  const float* kappa = (const float*)d_in[2];   // (64, 1)
  const float* pi    = (const float*)d_in[3];   // (64,)

  const int n_rows = in_sizes[0] / NP;

  char*  ws       = (char*)d_ws;
  float* mu_t     = (float*)ws;                         // 64 KB
  float* c0       = (float*)(ws + NP * NK * sizeof(float));
  float* c1       = c0 + NK;
  float* partials = c1 + NK;

  const int nblocks = (n_rows + ROWS_PER_BLOCK - 1) / ROWS_PER_BLOCK;

  watson_prep  <<<1,       THREADS, 0, stream>>>(mu, kappa, pi, mu_t, c0, c1);
  watson_main  <<<nblocks, THREADS, 0, stream>>>(X, mu_t, c0, c1, partials, n_rows);
  watson_reduce<<<1,       THREADS, 0, stream>>>(partials, nblocks, (float*)d_out);
}